// VectorQuantizer_19129784336699
// MI455X (gfx1250) — compile-verified
//
#include <hip/hip_runtime.h>
#include <stdint.h>

typedef __attribute__((ext_vector_type(2))) float v2f;
typedef __attribute__((ext_vector_type(8))) float v8f;

#define EDIM      256
#define NCODE     1024
#define TDIM      2048
#define NTOK      65536            // 32 * 2048
#define ZQ_ELEMS  16777216         // 32 * 256 * 2048
#define LOSS_OFF  ZQ_ELEMS
#define PERP_OFF  (ZQ_ELEMS + 1)
#define IDX_OFF   (ZQ_ELEMS + 2)

#define WS_LOSS   0
#define WS_CNT    1                // ws[1 .. 1024] : per-code counts
#define WS_ENORM  2048             // ws[2048 .. 3071] : ||e_j||^2

#define BS_STRIDE 260              // padded row stride (floats): conflict-free b64, 16B-aligned rows

// ---------------------------------------------------------------------------
// Kernel 1: per-code squared norms + zero the accumulators (ws is poisoned).
// ---------------------------------------------------------------------------
__global__ __launch_bounds__(256) void vq_prep(const float* __restrict__ E,
                                               float* __restrict__ ws) {
    int g = blockIdx.x * 256 + threadIdx.x;        // 0..1023 (grid = 4 blocks)
    const float4* row = (const float4*)(E + (size_t)g * EDIM);
    float s = 0.f;
#pragma unroll
    for (int i = 0; i < EDIM / 4; ++i) {
        float4 v = row[i];
        s += v.x * v.x + v.y * v.y + v.z * v.z + v.w * v.w;
    }
    ws[WS_ENORM + g] = s;
    ws[WS_CNT + g]   = 0.f;
    if (g == 0) ws[WS_LOSS] = 0.f;
}

// ---------------------------------------------------------------------------
// Kernel 2: fp32 WMMA distance GEMM + argmin + gather + loss/count partials.
// Block = 256 threads (8 waves), owns 128 token rows. Wave w owns 16 rows,
// A tile in 128 VGPRs. B tiles double-buffered in LDS via async global->LDS
// b128 copies (ASYNCcnt), 4 independent WMMA accumulator chains.
// ---------------------------------------------------------------------------
__global__ __launch_bounds__(256) void vq_main(const float* __restrict__ z,
                                               const float* __restrict__ E,
                                               float* __restrict__ ws,
                                               float* __restrict__ out) {
    __shared__ float Bs[2][16 * BS_STRIDE];   // double-buffered 16-code tiles
    __shared__ float enorm_s[NCODE];
    __shared__ int   idx_s[128];
    __shared__ float wsum[8];

    const int tid  = threadIdx.x;
    const int wave = tid >> 5;
    const int lane = tid & 31;
    const int l16  = lane & 15;
    const int hi   = lane >> 4;                    // 0: rows v, K+0/1 ; 1: rows v+8, K+2/3

    const int n0 = blockIdx.x * 128;               // first flat token of this block
    const int b  = n0 >> 11;                       // batch index (2048 t per batch)
    const int t0 = n0 & 2047;

    const uint64_t Ebase = (uint64_t)(uintptr_t)E;

    // stage all code norms into LDS (consumed after first in-loop barrier)
#pragma unroll
    for (int k = 0; k < 4; ++k)
        enorm_s[tid + k * 256] = ws[WS_ENORM + tid + k * 256];

    // ---- async prefetch of one 16-code B tile into Bs[jtn & 1] ----
    // per thread: 4 x b128 (16 B) memory->LDS, tracked by ASYNCcnt
    auto prefetch = [&](int jtn) {
        const int sel = jtn & 1;
#pragma unroll
        for (int it = 0; it < 4; ++it) {
            int i  = tid + it * 256;               // float4 index 0..1023
            int k  = i >> 6;                       // code row 0..15
            int c4 = i & 63;                       // float4 column
            uint32_t lds_addr = (uint32_t)(uintptr_t)&Bs[sel][k * BS_STRIDE + c4 * 4];
            uint32_t goff     = (uint32_t)(((jtn * 16 + k) * EDIM + c4 * 4) * 4);
            asm volatile("global_load_async_to_lds_b128 %0, %1, %2"
                         :: "v"(lds_addr), "v"(goff), "s"(Ebase) : "memory");
        }
    };

    // ---- A tile: 16 token rows x 256 dims, striped per WMMA f32 16x16x4 layout ----
    // zf[n][c] lives at z[b*EDIM*TDIM + c*TDIM + t]; lanes stride t -> coalesced.
    const int t_row = t0 + wave * 16 + l16;
    const float* zb = z + (size_t)b * (EDIM * TDIM) + t_row;
    v2f a[64];
#pragma unroll
    for (int kc = 0; kc < 64; ++kc) {
        int c0 = kc * 4 + (hi << 1);
        a[kc].x = zb[(size_t)c0 * TDIM];
        a[kc].y = zb[(size_t)(c0 + 1) * TDIM];
    }

    float best_val[8];
    int   best_idx[8];
#pragma unroll
    for (int v = 0; v < 8; ++v) { best_val[v] = 3.4e38f; best_idx[v] = 0; }

    prefetch(0);

    for (int jt = 0; jt < 64; ++jt) {
        // overlap: issue next tile's async copy, then drain only the current tile's 4 ops
        if (jt < 63) {
            prefetch(jt + 1);
            asm volatile("s_wait_asynccnt 0x4" ::: "memory");
        } else {
            asm volatile("s_wait_asynccnt 0x0" ::: "memory");
        }
        __syncthreads();                           // tile jt visible from all waves

        const float* bsl = &Bs[jt & 1][0] + l16 * BS_STRIDE + (hi << 1);

        // 4 independent accumulator chains over K = 256 (4 WMMAs always in flight)
        v8f acc0 = {}, acc1 = {}, acc2 = {}, acc3 = {};
#pragma unroll
        for (int kc = 0; kc < 16; ++kc) {
            v2f b0 = *(const v2f*)(bsl + (kc     ) * 4);
            v2f b1 = *(const v2f*)(bsl + (kc + 16) * 4);
            v2f b2 = *(const v2f*)(bsl + (kc + 32) * 4);
            v2f b3 = *(const v2f*)(bsl + (kc + 48) * 4);
            acc0 = __builtin_amdgcn_wmma_f32_16x16x4_f32(false, a[kc     ], false, b0, (short)0, acc0, false, false);
            acc1 = __builtin_amdgcn_wmma_f32_16x16x4_f32(false, a[kc + 16], false, b1, (short)0, acc1, false, false);
            acc2 = __builtin_amdgcn_wmma_f32_16x16x4_f32(false, a[kc + 32], false, b2, (short)0, acc2, false, false);
            acc3 = __builtin_amdgcn_wmma_f32_16x16x4_f32(false, a[kc + 48], false, b3, (short)0, acc3, false, false);
        }

        // distance proxy: ||e||^2 - 2 * (z . e)   (||z||^2 constant per row)
        int   j  = jt * 16 + l16;
        float en = enorm_s[j];
#pragma unroll
        for (int v = 0; v < 8; ++v) {
            float d = en - 2.0f * ((acc0[v] + acc1[v]) + (acc2[v] + acc3[v]));
            if (d < best_val[v]) { best_val[v] = d; best_idx[v] = j; }
        }
        __syncthreads();                           // all waves done reading Bs[jt&1]
    }

    // reduce argmin across the 16 lanes holding the same row (xor within half-wave)
#pragma unroll
    for (int off = 8; off >= 1; off >>= 1) {
#pragma unroll
        for (int v = 0; v < 8; ++v) {
            float ov = __shfl_xor(best_val[v], off, 32);
            int   oi = __shfl_xor(best_idx[v], off, 32);
            if (ov < best_val[v] || (ov == best_val[v] && oi < best_idx[v])) {
                best_val[v] = ov; best_idx[v] = oi;
            }
        }
    }

    // lanes 0 / 16 hold rows (wave*16 + hi*8 + v); publish idx
    if (l16 == 0) {
#pragma unroll
        for (int v = 0; v < 8; ++v) {
            int r = wave * 16 + (hi << 3) + v;      // local row 0..127
            idx_s[r] = best_idx[v];
            out[(size_t)IDX_OFF + n0 + r] = (float)best_idx[v];
        }
    }
    __syncthreads();

    if (tid < 128) atomicAdd(&ws[WS_CNT + idx_s[tid]], 1.0f);

    // gather z_q, write [B, C, T] layout (t-coalesced), accumulate loss partial
    float lsum = 0.f;
    const size_t obase = (size_t)b * (EDIM * TDIM) + t0;
    for (int i = tid; i < EDIM * 128; i += 256) {
        int c  = i >> 7;
        int tl = i & 127;
        int j  = idx_s[tl];
        float e  = E[(size_t)j * EDIM + c];
        size_t off = obase + (size_t)c * TDIM + tl;
        float zv = z[off];
        float df = e - zv;
        lsum += df * df;
        out[off] = e;                               // z_q_st == z_q numerically
    }

#pragma unroll
    for (int off = 16; off >= 1; off >>= 1) lsum += __shfl_xor(lsum, off, 32);
    if (lane == 0) wsum[wave] = lsum;
    __syncthreads();
    if (tid == 0) {
        float s = 0.f;
#pragma unroll
        for (int w = 0; w < 8; ++w) s += wsum[w];
        atomicAdd(&ws[WS_LOSS], s);
    }
}

// ---------------------------------------------------------------------------
// Kernel 3: loss scalar + perplexity from counts.
// ---------------------------------------------------------------------------
__global__ __launch_bounds__(256) void vq_finalize(const float* __restrict__ ws,
                                                   float* __restrict__ out) {
    __shared__ float hs[256];
    int tid = threadIdx.x;
    float h = 0.f;
#pragma unroll
    for (int k = 0; k < 4; ++k) {
        float cnt = ws[WS_CNT + tid + k * 256];
        float em  = cnt * (1.0f / (float)NTOK);
        h += em * __logf(em + 1e-10f);
    }
    hs[tid] = h;
    __syncthreads();
    for (int s = 128; s > 0; s >>= 1) {
        if (tid < s) hs[tid] += hs[tid + s];
        __syncthreads();
    }
    if (tid == 0) {
        // loss = beta*mean + mean = 1.25 * mean((z_q - z)^2)
        out[LOSS_OFF] = 1.25f * ws[WS_LOSS] / (float)ZQ_ELEMS;
        out[PERP_OFF] = __expf(-hs[0]);
    }
}

extern "C" void kernel_launch(void* const* d_in, const int* in_sizes, int n_in,
                              void* d_out, int out_size, void* d_ws, size_t ws_size,
                              hipStream_t stream) {
    const float* z  = (const float*)d_in[0];   // [32, 256, 2048] fp32
    const float* E  = (const float*)d_in[1];   // [1024, 256] fp32
    float* out = (float*)d_out;
    float* ws  = (float*)d_ws;

    vq_prep<<<dim3(4),   dim3(256), 0, stream>>>(E, ws);
    vq_main<<<dim3(512), dim3(256), 0, stream>>>(z, E, ws, out);
    vq_finalize<<<dim3(1), dim3(256), 0, stream>>>(ws, out);
}